// Model_66589172957779
// MI455X (gfx1250) — compile-verified
//
#include <hip/hip_runtime.h>

// ---------------------------------------------------------------------------
// GRU(T=262144, H=128, F=8) + dense head, for MI455X (gfx1250, wave32).
//
// Kernel 1 (gru_scan_kernel): the inherently serial recurrence. One workgroup,
//   384 threads (one per gate column of 3H), entire 196KB recurrent kernel
//   staged in LDS (CDNA5 WGP has 320KB LDS). Input projection (8 FMAs) fused.
//   Writes hs[T,128] to workspace.
// Kernel 2 (dense_wmma_kernel): hs @ dense_kernel + bias using
//   V_WMMA_F32_16X16X4_F32. 6.4 GFLOP over ~234MB traffic -> memory bound at
//   23.3 TB/s. B weights stored k-pair-interleaved in LDS so each fragment is
//   one ds_load_b64 into an aligned VGPR pair (no repacking movs).
// ---------------------------------------------------------------------------

typedef __attribute__((ext_vector_type(2))) float v2f;
typedef __attribute__((ext_vector_type(8))) float v8f;

#define Hh   128   // GRU units
#define Ff   8     // input features per step
#define G3   384   // 3*H gate columns
#define NOUT 96    // dense output width

__device__ __forceinline__ float sigmoid_fast(float x) {
    return 1.0f / (1.0f + __expf(-x));
}

// ------------------------- Kernel 1: sequential GRU scan -------------------
__global__ __launch_bounds__(G3) void gru_scan_kernel(
    const float* __restrict__ even_x,          // [T, 8]
    const float* __restrict__ gru_kernel,      // [8, 384]
    const float* __restrict__ gru_rec_kernel,  // [128, 384]
    const float* __restrict__ gru_bias,        // [2, 384]
    float* __restrict__ hs,                    // [T, 128] (workspace)
    int T)
{
    extern __shared__ float smem[];
    float* Wr  = smem;                    // [128*384] recurrent kernel (196KB)
    float* Wk  = Wr  + Hh * G3;           // [8*384]   input kernel
    float* hsh = Wk  + Ff * G3;           // [128]     hidden state
    float* sa  = hsh + Hh;                // [384]     gate pre-activations
    float* shn = sa  + G3;                // [128]     recurrent part of h-gate
    float* sx  = shn + Hh;                // [8]       current (shifted) input

    const int j = threadIdx.x;            // gate column 0..383

    // Stage weights into LDS (coalesced, conflict-free).
    for (int i = j; i < Hh * G3; i += G3) Wr[i] = gru_rec_kernel[i];
    for (int i = j; i < Ff * G3; i += G3) Wk[i] = gru_kernel[i];
    if (j < Hh) hsh[j] = 0.0f;

    const float bi = gru_bias[j];         // input bias (row 0)
    const float br = gru_bias[G3 + j];    // recurrent bias (row 1)

    for (int t = 0; t < T; ++t) {
        // x is shifted right by one step (zeros at t==0).
        if (j < Ff) sx[j] = (t == 0) ? 0.0f : even_x[(size_t)(t - 1) * Ff + j];
        __syncthreads();   // x ready; previous h_new visible

        float gx = bi;
        #pragma unroll
        for (int k = 0; k < Ff; ++k) gx = fmaf(sx[k], Wk[k * G3 + j], gx);

        float gh = br;
        #pragma unroll 8
        for (int k = 0; k < Hh; ++k) gh = fmaf(hsh[k], Wr[k * G3 + j], gh);

        if (j < 2 * Hh) {
            sa[j] = gx + gh;              // z and r gates: x-part + h-part
        } else {
            sa[j]           = gx;         // h-candidate: keep parts separate
            shn[j - 2 * Hh] = gh;         // (reset gate scales only gh)
        }
        __syncthreads();   // gates ready; all reads of hsh done

        if (j < Hh) {
            float z  = sigmoid_fast(sa[j]);
            float r  = sigmoid_fast(sa[Hh + j]);
            float hh = tanhf(fmaf(r, shn[j], sa[2 * Hh + j]));
            float hn = fmaf(z, hsh[j] - hh, hh);   // z*h + (1-z)*hh
            hsh[j] = hn;
            hs[(size_t)t * Hh + j] = hn;
        }
        // barrier at top of next iteration protects hsh/sx reuse
    }
}

// ---------------- Kernel 2: hs @ dense_kernel + bias via WMMA --------------
// One wave per 16-row x 96-col output panel; 6 f32 16x16 accumulator tiles;
// K=128 consumed 4 at a time by V_WMMA_F32_16X16X4_F32.
// LDS holds Wd k-pair interleaved: wlds[p][n] = {Wd[2p][n], Wd[2p+1][n]},
// so a B fragment is a single aligned ds_load_b64 per tile.
__global__ __launch_bounds__(256) void dense_wmma_kernel(
    const float* __restrict__ hs,    // [T, 128]
    const float* __restrict__ Wd,    // [128, 96]
    const float* __restrict__ bd,    // [96]
    float* __restrict__ out,         // [T, 96]
    int T)
{
    __shared__ v2f wlds[(Hh / 2) * NOUT];          // 64 pairs x 96 cols = 48 KB
    for (int i = threadIdx.x; i < (Hh / 2) * NOUT; i += 256) {
        const int p = i / NOUT;                    // k-pair index 0..63
        const int n = i - p * NOUT;                // column 0..95
        v2f w;
        w.x = Wd[(2 * p) * NOUT + n];
        w.y = Wd[(2 * p + 1) * NOUT + n];
        wlds[i] = w;
    }
    __syncthreads();

    const int lane = threadIdx.x & 31;
    const int wave = threadIdx.x >> 5;             // 0..7
    const int half = lane >> 4;                    // 0: K pair +0, 1: K pair +1
    const int l15  = lane & 15;
    const long row0 = (long)blockIdx.x * 128 + (long)wave * 16;

    // C/D layout: VGPR r -> row (r + 8*half), col l15. Bias depends on col only.
    v8f acc[6];
    #pragma unroll
    for (int nt = 0; nt < 6; ++nt) {
        float b = bd[nt * 16 + l15];
        #pragma unroll
        for (int r = 0; r < 8; ++r) acc[nt][r] = b;
    }

    long arow = row0 + l15;                        // A row owned by this lane
    if (arow >= T) arow = T - 1;                   // clamp (stores are guarded)
    const float* aptr = hs + arow * Hh + half * 2; // lane's K pair within group of 4

    // Per-lane base into the pair-interleaved B matrix: pair offset (half) and
    // column (l15) are lane-constant; per-iteration offsets become immediates.
    const v2f* bptr = wlds + (size_t)half * NOUT + l15;

    for (int kk = 0; kk < Hh; kk += 4) {
        // A 16x4 f32 fragment (ISA 7.12.2): lanes 0-15 hold K=kk,kk+1;
        // lanes 16-31 hold K=kk+2,kk+3 for the same 16 rows. Single b64 load.
        v2f a = *(const v2f*)(aptr + kk);
        const int pbase = (kk >> 1) * NOUT;        // k-pair row base in wlds
        #pragma unroll
        for (int nt = 0; nt < 6; ++nt) {
            v2f b = bptr[pbase + nt * 16];         // one ds_load_b64
            acc[nt] = __builtin_amdgcn_wmma_f32_16x16x4_f32(
                /*neg_a=*/false, a, /*neg_b=*/false, b,
                /*c_mod=*/(short)0, acc[nt], /*reuse_a=*/false, /*reuse_b=*/false);
        }
    }

    #pragma unroll
    for (int nt = 0; nt < 6; ++nt) {
        #pragma unroll
        for (int r = 0; r < 8; ++r) {
            long trow = row0 + r + half * 8;
            if (trow < T) out[trow * NOUT + nt * 16 + l15] = acc[nt][r];
        }
    }
}

// --------------------------------- host ------------------------------------
extern "C" void kernel_launch(void* const* d_in, const int* in_sizes, int n_in,
                              void* d_out, int out_size, void* d_ws, size_t ws_size,
                              hipStream_t stream) {
    const float* even_x = (const float*)d_in[0];   // [T,1,1,8]
    const float* gk     = (const float*)d_in[1];   // [8,384]
    const float* grk    = (const float*)d_in[2];   // [128,384]
    const float* gb     = (const float*)d_in[3];   // [2,384]
    const float* dk     = (const float*)d_in[4];   // [128,96]
    const float* db     = (const float*)d_in[5];   // [96]
    float* out = (float*)d_out;

    const int T = in_sizes[0] / Ff;
    float* hs = (float*)d_ws;                      // needs T*128*4 = 128 MB

    const size_t scanLds =
        (size_t)(Hh * G3 + Ff * G3 + Hh + G3 + Hh + Ff) * sizeof(float); // ~207 KB

    // Allow >64KB dynamic LDS for the scan kernel (CDNA5 WGP: 320KB).
    // Deterministic/idempotent every call; return value intentionally ignored.
    (void)hipFuncSetAttribute(reinterpret_cast<const void*>(gru_scan_kernel),
                              hipFuncAttributeMaxDynamicSharedMemorySize,
                              (int)scanLds);

    gru_scan_kernel<<<dim3(1), dim3(G3), scanLds, stream>>>(
        even_x, gk, grk, gb, hs, T);

    const int blocks = (T + 127) / 128;
    dense_wmma_kernel<<<dim3(blocks), dim3(256), 0, stream>>>(
        hs, dk, db, out, T);
}